// VGCN_28346784154176
// MI455X (gfx1250) — compile-verified
//
#include <hip/hip_runtime.h>
#include <hip/hip_bf16.h>

typedef __attribute__((ext_vector_type(2))) float v2f;
typedef __attribute__((ext_vector_type(8))) float v8f;
typedef int i32x4 __attribute__((vector_size(16)));
typedef int i32x2 __attribute__((vector_size(8)));

#define BM 64
#define BN 32
#define BK 16
#define APAD 4   // A LDS row stride = 20 floats = 80B (16B multiple for b128)
#define BPAD 2   // B LDS row stride = 34 floats = 136B (8B multiple for b64)

#define GAS __attribute__((address_space(1)))
#define LAS __attribute__((address_space(3)))

// ---- CDNA5 async global->LDS copies (ASYNCcnt-tracked, no VGPR round-trip) --
__device__ __forceinline__ void async_ld_b128(void* lds, const void* gsrc) {
#if __has_builtin(__builtin_amdgcn_global_load_async_to_lds_b128)
    __builtin_amdgcn_global_load_async_to_lds_b128(
        (GAS i32x4*)gsrc, (LAS i32x4*)lds, 0, 0);
#else
    unsigned l = (unsigned)(uintptr_t)(LAS void*)lds;
    asm volatile("global_load_async_to_lds_b128 %0, %1, off"
                 :: "v"(l), "v"((unsigned long long)(uintptr_t)gsrc) : "memory");
#endif
}
__device__ __forceinline__ void async_ld_b64(void* lds, const void* gsrc) {
#if __has_builtin(__builtin_amdgcn_global_load_async_to_lds_b64)
    __builtin_amdgcn_global_load_async_to_lds_b64(
        (GAS i32x2*)gsrc, (LAS i32x2*)lds, 0, 0);
#else
    unsigned l = (unsigned)(uintptr_t)(LAS void*)lds;
    asm volatile("global_load_async_to_lds_b64 %0, %1, off"
                 :: "v"(l), "v"((unsigned long long)(uintptr_t)gsrc) : "memory");
#endif
}
template<int Ncnt>
__device__ __forceinline__ void wait_async() {
#if __has_builtin(__builtin_amdgcn_s_wait_asynccnt)
    __builtin_amdgcn_s_wait_asynccnt(Ncnt);
#else
    asm volatile("s_wait_asynccnt %0" :: "i"(Ncnt));
#endif
}

// ---------------------------------------------------------------------------
// fp32 WMMA GEMM with double-buffered async-LDS staging:
//   C[M,Nd] = act( [A0|A1] @ B + bias ),  split at column K0.
// 256 threads = 8 waves; wave (wm 0..3, wn 0..1) owns a 16x16 C tile.
// Every thread issues exactly 2 async ops per K-stage => per-wave ASYNCcnt
// is uniform and s_wait_asynccnt 2 pipelines one stage ahead.
// ---------------------------------------------------------------------------
__global__ __launch_bounds__(256) void gemm_f32_wmma(
    const float* __restrict__ A0, const float* __restrict__ A1, int K0, int K,
    const float* __restrict__ B, const float* __restrict__ bias,
    float* __restrict__ C, int M, int Nd, int doRelu)
{
    __shared__ float Alds[2][BM][BK + APAD];
    __shared__ float Blds[2][BK][BN + BPAD];
    __shared__ __align__(16) float Trash[8];

    const int tid  = threadIdx.x;
    const int lane = tid & 31;
    const int wave = tid >> 5;      // 0..7
    const int wm   = wave & 3;      // M tile (4)
    const int wn   = wave >> 2;     // N tile (2)
    const int lr   = lane & 15;
    const int lh   = lane >> 4;

    const int rowBase = blockIdx.x * BM;
    const int colBase = blockIdx.y * BN;
    const int lda1 = K - K0;

    // stage-invariant per-thread staging coordinates
    const int ar = (tid * 4) >> 4, ak = (tid * 4) & 15;   // A: 4 floats (16B)
    const int br = (tid * 2) >> 5, bc = (tid * 2) & 31;   // B: 2 floats (8B)
    const int grow = rowBase + ar;
    const int gcol = colBase + bc;
    const bool aOk = grow < M;
    const bool bOk = (gcol + 1) < Nd;          // Nd even for all call sites

    // LDS slots never async-filled (OOB) are zeroed once; they stay zero.
    if (!aOk) {
        #pragma unroll
        for (int b = 0; b < 2; ++b)
            #pragma unroll
            for (int q = 0; q < 4; ++q) Alds[b][ar][ak + q] = 0.f;
    }
    if (!bOk) {
        #pragma unroll
        for (int b = 0; b < 2; ++b) {
            Blds[b][br][bc] = 0.f;
            Blds[b][br][bc + 1] = 0.f;
        }
    }

    v8f acc = {};
    const int S = K / BK;           // K is a multiple of BK at all call sites

    auto issue = [&](int s, int bufi) {
        const int gka = s * BK + ak;
        const float* asrc =
            aOk ? ((gka < K0) ? (A0 + (size_t)grow * K0 + gka)
                              : (A1 + (size_t)grow * lda1 + (gka - K0)))
                : A0;                                   // valid dummy
        void* adst = aOk ? (void*)&Alds[bufi][ar][ak] : (void*)&Trash[0];
        async_ld_b128(adst, asrc);

        const int gkb = s * BK + br;
        const float* bsrc = bOk ? (B + (size_t)gkb * Nd + gcol) : B;
        void* bdst = bOk ? (void*)&Blds[bufi][br][bc] : (void*)&Trash[4];
        async_ld_b64(bdst, bsrc);
    };

    issue(0, 0);
    int buf = 0;
    for (int s = 0; s < S; ++s) {
        if (s + 1 < S) {
            issue(s + 1, buf ^ 1);   // prefetch next stage into other buffer
            wait_async<2>();         // previous stage's 2 ops complete
        } else {
            wait_async<0>();
        }
        __syncthreads();

        #pragma unroll
        for (int kk = 0; kk < BK; kk += 4) {
            const int k2 = kk + 2 * lh;   // lane-half selects K pair
            v2f a, b;
            a.x = Alds[buf][wm * 16 + lr][k2 + 0];
            a.y = Alds[buf][wm * 16 + lr][k2 + 1];
            b.x = Blds[buf][k2 + 0][wn * 16 + lr];
            b.y = Blds[buf][k2 + 1][wn * 16 + lr];
            acc = __builtin_amdgcn_wmma_f32_16x16x4_f32(
                      /*neg_a=*/false, a, /*neg_b=*/false, b,
                      /*c_mod=*/(short)0, acc,
                      /*reuse_a=*/false, /*reuse_b=*/false);
        }
        __syncthreads();
        buf ^= 1;
    }

    // C 16x16 f32 layout: VGPR j -> row (j + 8*lh), col lr
    #pragma unroll
    for (int j = 0; j < 8; ++j) {
        int r = rowBase + wm * 16 + j + 8 * lh;
        int c = colBase + wn * 16 + lr;
        if (r < M && c < Nd) {
            float v = acc[j];
            if (bias)   v += bias[c];
            if (doRelu) v = fmaxf(v, 0.f);
            C[(size_t)r * Nd + c] = v;
        }
    }
}

// ---------------------------------------------------------------------------
__global__ void fill_zero(float* __restrict__ p, long n)
{
    long i = (long)blockIdx.x * blockDim.x + threadIdx.x;
    long stride = (long)gridDim.x * blockDim.x;
    for (; i < n; i += stride) p[i] = 0.f;
}

// ---------------------------------------------------------------------------
// Edge scatter: acc[dst] += support[src] * w.  One wave32 per edge;
// unsafeAtomicAdd -> native global_atomic_add_f32 (no CAS loop).
// ---------------------------------------------------------------------------
__global__ __launch_bounds__(256) void spmm_scatter(
    const float* __restrict__ support, const int* __restrict__ src,
    const int* __restrict__ dst, const float* __restrict__ w,
    float* __restrict__ acc, int E, int D)
{
    int e = (int)(((long)blockIdx.x * blockDim.x + threadIdx.x) >> 5);
    if (e >= E) return;
    int lane = threadIdx.x & 31;
    int s = src[e];
    int d = dst[e];
    float ew = w[e];
    const float* sp = support + (size_t)s * D;
    float*       ap = acc     + (size_t)d * D;
    for (int f = lane; f < D; f += 32)
        unsafeAtomicAdd(&ap[f], sp[f] * ew);
}

// ---------------------------------------------------------------------------
__global__ void bias_act(float* __restrict__ buf, const float* __restrict__ bias,
                         long n, int D, int doRelu)
{
    long i = (long)blockIdx.x * blockDim.x + threadIdx.x;
    if (i >= n) return;
    float v = buf[i] + bias[(int)(i % D)];
    if (doRelu) v = fmaxf(v, 0.f);
    buf[i] = v;
}

// ---------------------------------------------------------------------------
__global__ void reparam(const float* __restrict__ mu, const float* __restrict__ lv,
                        const float* __restrict__ eps, float* __restrict__ z, long n)
{
    long i = (long)blockIdx.x * blockDim.x + threadIdx.x;
    if (i >= n) return;
    z[i] = mu[i] + eps[i] * __expf(lv[i]);
}

// ---------------------------------------------------------------------------
__global__ void log_softmax40(const float* __restrict__ acc,
                              const float* __restrict__ b2,
                              float* __restrict__ out, int N)
{
    int r = blockIdx.x * blockDim.x + threadIdx.x;
    if (r >= N) return;
    const float* a = acc + (size_t)r * 40;
    float m = -3.402823466e+38f;
    #pragma unroll
    for (int c = 0; c < 40; ++c) m = fmaxf(m, a[c] + b2[c]);
    float s = 0.f;
    #pragma unroll
    for (int c = 0; c < 40; ++c) s += __expf(a[c] + b2[c] - m);
    float ls = __logf(s);
    float* o = out + (size_t)r * 40;
    #pragma unroll
    for (int c = 0; c < 40; ++c) o[c] = a[c] + b2[c] - m - ls;
}

// ---------------------------------------------------------------------------
extern "C" void kernel_launch(void* const* d_in, const int* in_sizes, int n_in,
                              void* d_out, int out_size, void* d_ws, size_t ws_size,
                              hipStream_t stream)
{
    const int N = 50000, E = 800000;
    const int NFEAT = 512, NHID = 256, NCODE = 64, NCLASS = 40;

    const float* x    = (const float*)d_in[0];
    const int*   esrc = (const int*)  d_in[1];
    const int*   edst = (const int*)  d_in[2];
    const float* ew   = (const float*)d_in[3];
    const float* eps  = (const float*)d_in[4];
    const float* W1   = (const float*)d_in[5];
    const float* b1   = (const float*)d_in[6];
    const float* Wmu  = (const float*)d_in[7];
    const float* bmu  = (const float*)d_in[8];
    const float* Wlv  = (const float*)d_in[9];
    const float* blv  = (const float*)d_in[10];
    const float* Wdec = (const float*)d_in[11];
    const float* bdec = (const float*)d_in[12];
    const float* W2   = (const float*)d_in[13];
    const float* b2   = (const float*)d_in[14];
    float* out = (float*)d_out;

    // workspace layout (fp32)
    float* s1 = (float*)d_ws;                    // [N, 256] support1 = x @ W1
    float* h1 = s1 + (size_t)N * NHID;           // [N, 256] acc -> relu(acc+b1)
    float* mu = h1 + (size_t)N * NHID;           // [N, 64]
    float* lv = mu + (size_t)N * NCODE;          // [N, 64]
    float* z  = lv + (size_t)N * NCODE;          // [N, 64]
    float* x1 = z  + (size_t)N * NCODE;          // [N, 256]
    float* s2 = x1 + (size_t)N * NHID;           // [N, 40]  support2
    float* a2 = s2 + (size_t)N * NCLASS;         // [N, 40]  scatter acc

    const dim3 blk(256);
    const int gM = (N + BM - 1) / BM;            // 782 row blocks

    // 1. s1 = x @ W1          (bias applied AFTER aggregation per reference)
    gemm_f32_wmma<<<dim3(gM, NHID / BN), blk, 0, stream>>>(
        x, x, NFEAT, NFEAT, W1, nullptr, s1, N, NHID, 0);

    // 2-4. h1 = relu(spmm(adj, s1) + b1)
    fill_zero<<<dim3(20000), blk, 0, stream>>>(h1, (long)N * NHID);
    spmm_scatter<<<dim3((E * 32 + 255) / 256), blk, 0, stream>>>(
        s1, esrc, edst, ew, h1, E, NHID);
    bias_act<<<dim3(((long)N * NHID + 255) / 256), blk, 0, stream>>>(
        h1, b1, (long)N * NHID, NHID, 1);

    // 5-6. mu / log_var
    gemm_f32_wmma<<<dim3(gM, (NCODE + BN - 1) / BN), blk, 0, stream>>>(
        h1, h1, NHID, NHID, Wmu, bmu, mu, N, NCODE, 0);
    gemm_f32_wmma<<<dim3(gM, (NCODE + BN - 1) / BN), blk, 0, stream>>>(
        h1, h1, NHID, NHID, Wlv, blv, lv, N, NCODE, 0);

    // 7. z = mu + eps * exp(log_var)
    reparam<<<dim3(((long)N * NCODE + 255) / 256), blk, 0, stream>>>(
        mu, lv, eps, z, (long)N * NCODE);

    // 8. x1 = relu(z @ Wdec + bdec)
    gemm_f32_wmma<<<dim3(gM, NHID / BN), blk, 0, stream>>>(
        z, z, NCODE, NCODE, Wdec, bdec, x1, N, NHID, 1);

    // 9. s2 = concat([x1, h1], 1) @ W2   (split-A, no concat materialized)
    gemm_f32_wmma<<<dim3(gM, (NCLASS + BN - 1) / BN), blk, 0, stream>>>(
        x1, h1, NHID, 2 * NHID, W2, nullptr, s2, N, NCLASS, 0);

    // 10-11. a2 = spmm(adj, s2)
    fill_zero<<<dim3(8000), blk, 0, stream>>>(a2, (long)N * NCLASS);
    spmm_scatter<<<dim3((E * 32 + 255) / 256), blk, 0, stream>>>(
        s2, esrc, edst, ew, a2, E, NCLASS);

    // 12. out = log_softmax(a2 + b2)
    log_softmax40<<<dim3((N + 255) / 256), blk, 0, stream>>>(a2, b2, out, N);
}